// GraphMultiHeadAttLayer_66168266162400
// MI455X (gfx1250) — compile-verified
//
#include <hip/hip_runtime.h>
#include <math.h>

#define FDIM 128
#define HEADS 4
#define ALPHA 0.2f

typedef float v2f __attribute__((ext_vector_type(2)));
typedef float v8f __attribute__((ext_vector_type(8)));

// ---------------------------------------------------------------------------
// Kernel 1: build fused projection matrix B [FDIM][8]:
//   B[f][k]   = W[k][f] * a[k][f]        (src-score column, heads 0..3)
//   B[f][4+k] = W[k][f] * a[k][FDIM+f]   (dst-score column, heads 0..3)
// ---------------------------------------------------------------------------
__global__ void gat_build_B(const float* __restrict__ W, const float* __restrict__ a,
                            float* __restrict__ B) {
  int f = threadIdx.x;  // launched with 128 threads
  if (f < FDIM) {
#pragma unroll
    for (int k = 0; k < HEADS; ++k) {
      float w = W[k * FDIM + f];
      B[f * 8 + k]     = w * a[k * 2 * FDIM + f];
      B[f * 8 + 4 + k] = w * a[k * 2 * FDIM + FDIM + f];
    }
  }
}

// ---------------------------------------------------------------------------
// Kernel 2: scores = inputs(N x 128) @ B(128 x 8) via V_WMMA_F32_16X16X4_F32.
// One wave32 per 16-row tile; K=128 swept in 32 steps of K=4.
// B is staged into LDS zero-padded to [128][16] so the operand load in the
// K-loop is an unconditional ds_load (no exec-mask churn between WMMAs).
// The 16x8 C tile is transposed through LDS and stored as one coalesced
// 512B burst (one float4 per lane) instead of scattered b32 stores.
// A 16x4 f32 layout: lane L -> M=L%16, VGPR0=K(2*(L/16)), VGPR1=K(+1).
// B 4x16 f32 layout: lane L -> N=L%16, VGPR0=K(2*(L/16)), VGPR1=K(+1).
// C/D:   VGPR v, lanes 0-15 -> M=v, lanes 16-31 -> M=v+8, N=lane%16.
// ---------------------------------------------------------------------------
__global__ void gat_scores_wmma(const float* __restrict__ X, const float* __restrict__ B,
                                float* __restrict__ S, int N) {
  __shared__ float ldsB[FDIM * 16];  // zero-padded: cols 8..15 are 0
  __shared__ float ldsS[16 * 8];     // C-tile staging for coalesced store

  const int lane = threadIdx.x;      // 0..31 (blockDim.x == 32)

  // stage B -> LDS with zero padding (1024 real + 1024 zero floats)
  for (int i = lane; i < FDIM * 16; i += 32) {
    const int k = i >> 4;
    const int j = i & 15;
    ldsB[i] = (j < 8) ? B[k * 8 + j] : 0.0f;
  }
  __syncthreads();

  const int row0 = blockIdx.x * 16;
  const int m    = lane & 15;        // row within tile (A) / column (B)
  const int half = lane >> 4;        // selects K pair {0,1} vs {2,3}

  int arow = row0 + m;
  if (arow >= N) arow = N - 1;       // clamp load address; EXEC stays all-1s
  const float* xrow = X + (size_t)arow * FDIM;

  v8f c = {};
#pragma unroll 4
  for (int k0 = 0; k0 < FDIM; k0 += 4) {
    const int ka = k0 + 2 * half;
    v2f av, bv;
    av.x = xrow[ka];
    av.y = xrow[ka + 1];
    bv.x = ldsB[ka * 16 + m];
    bv.y = ldsB[(ka + 1) * 16 + m];
    c = __builtin_amdgcn_wmma_f32_16x16x4_f32(
        /*neg_a=*/false, av, /*neg_b=*/false, bv,
        /*c_mod=*/(short)0, c, /*reuse_a=*/false, /*reuse_b=*/false);
  }

  // transpose C through LDS: lane (m<8) owns column m for local rows
  // 8*half .. 8*half+7
  if (m < 8) {
    const int mb = 8 * half;
#pragma unroll
    for (int v = 0; v < 8; ++v) ldsS[(mb + v) * 8 + m] = c[v];
  }
  __syncthreads();

  // coalesced store: each lane writes one float4 (whole tile = 512B burst)
  {
    const int r  = lane >> 1;          // local row 0..15
    const int c4 = (lane & 1) * 4;     // col 0 or 4
    const int grow = row0 + r;
    if (grow < N) {
      const float4 val = *(const float4*)(ldsS + r * 8 + c4);
      *(float4*)(S + (size_t)grow * 8 + c4) = val;
    }
  }
}

// ---------------------------------------------------------------------------
// Kernel 3: per-edge attention weights for all heads + rowsum (segment over src).
//   e_k = exp(leaky_relu(s_src[s][k] + s_dst[d][k])) * adj[e]
// ---------------------------------------------------------------------------
__global__ void gat_edge_pass(const int* __restrict__ src, const int* __restrict__ dst,
                              const float* __restrict__ adj, const float* __restrict__ S,
                              float* __restrict__ e_ws, float* __restrict__ rowsum,
                              int N, int E) {
  const int e = blockIdx.x * blockDim.x + threadIdx.x;
  if (e >= E) return;
  const int s = src[e];
  const int d = dst[e];
  const float av = adj[e];
#pragma unroll
  for (int k = 0; k < HEADS; ++k) {
    float x  = S[(size_t)s * 8 + k] + S[(size_t)d * 8 + 4 + k];
    float lr = (x > 0.0f) ? x : ALPHA * x;
    float ek = __expf(lr) * av;
    e_ws[(size_t)k * E + e] = ek;
    atomicAdd(rowsum + (size_t)k * N + s, ek);
  }
}

// ---------------------------------------------------------------------------
// Kernel 4: scatter for one head. One wave32 per edge; each lane handles a
// float4 of features. Accumulates raw sum_e e_k * inputs[dst] into hbuf[src]
// (the per-head W scale is folded into the finalize kernel).
// ---------------------------------------------------------------------------
__global__ void gat_scatter_head(const int* __restrict__ src, const int* __restrict__ dst,
                                 const float* __restrict__ ek_arr,
                                 const float* __restrict__ X,
                                 float* __restrict__ hbuf, int E) {
  const int wave = (int)((blockIdx.x * (size_t)blockDim.x + threadIdx.x) >> 5);
  const int lane = threadIdx.x & 31;
  if (wave >= E) return;
  const int s   = src[wave];
  const int d   = dst[wave];
  const float e = ek_arr[wave];

  const float4 v = ((const float4*)(X + (size_t)d * FDIM))[lane];
  float* hb = hbuf + (size_t)s * FDIM + lane * 4;
  atomicAdd(hb + 0, e * v.x);
  atomicAdd(hb + 1, e * v.y);
  atomicAdd(hb + 2, e * v.z);
  atomicAdd(hb + 3, e * v.w);
}

// ---------------------------------------------------------------------------
// Kernel 5: finalize one head: out += 0.25 * elu(hbuf * W[k] / rowsum[k]).
// ---------------------------------------------------------------------------
__global__ void gat_finalize_head(const float* __restrict__ hbuf,
                                  const float* __restrict__ Wk,
                                  const float* __restrict__ rowsum_k,
                                  float* __restrict__ out, int N) {
  const int idx   = blockIdx.x * blockDim.x + threadIdx.x;  // one float4 each
  const int total = N * (FDIM / 4);
  if (idx >= total) return;
  const int n  = idx / (FDIM / 4);
  const int f4 = (idx % (FDIM / 4)) * 4;

  const float inv = 1.0f / rowsum_k[n];
  const float4 hv = *(const float4*)(hbuf + (size_t)n * FDIM + f4);
  const float4 wv = *(const float4*)(Wk + f4);
  float4 ov = *(float4*)(out + (size_t)n * FDIM + f4);

  float x;
  x = hv.x * wv.x * inv; ov.x += 0.25f * ((x > 0.0f) ? x : (__expf(x) - 1.0f));
  x = hv.y * wv.y * inv; ov.y += 0.25f * ((x > 0.0f) ? x : (__expf(x) - 1.0f));
  x = hv.z * wv.z * inv; ov.z += 0.25f * ((x > 0.0f) ? x : (__expf(x) - 1.0f));
  x = hv.w * wv.w * inv; ov.w += 0.25f * ((x > 0.0f) ? x : (__expf(x) - 1.0f));

  *(float4*)(out + (size_t)n * FDIM + f4) = ov;
}

// ---------------------------------------------------------------------------
// Launcher. Workspace layout (floats):
//   B        : 128*8            (4 KB)
//   scores S : N*8              (3.2 MB)
//   rowsum   : HEADS*N          (1.6 MB)
//   e_ws     : HEADS*E          (25.6 MB)
//   hbuf     : N*FDIM           (51.2 MB)   -> ~82 MB total
// ---------------------------------------------------------------------------
extern "C" void kernel_launch(void* const* d_in, const int* in_sizes, int n_in,
                              void* d_out, int out_size, void* d_ws, size_t ws_size,
                              hipStream_t stream) {
  const float* X   = (const float*)d_in[0];   // [N, 128]
  const int*   edg = (const int*)  d_in[1];   // [2, E]
  const float* adj = (const float*)d_in[2];   // [E]
  const float* W   = (const float*)d_in[3];   // [4, 128]
  const float* a   = (const float*)d_in[4];   // [4, 256]
  float* out = (float*)d_out;

  const int N = in_sizes[0] / FDIM;
  const int E = in_sizes[2];
  const int* src = edg;
  const int* dst = edg + E;

  float* ws     = (float*)d_ws;
  float* B      = ws;
  float* S      = B + FDIM * 8;
  float* rowsum = S + (size_t)N * 8;
  float* e_ws   = rowsum + (size_t)HEADS * N;
  float* hbuf   = e_ws + (size_t)HEADS * E;

  // 1. fused projection matrix
  gat_build_B<<<1, 128, 0, stream>>>(W, a, B);

  // 2. per-node scores via f32 WMMA (one wave per 16-row tile)
  gat_scores_wmma<<<(N + 15) / 16, 32, 0, stream>>>(X, B, S, N);

  // 3. edge attention weights + rowsums (all heads in one pass)
  hipMemsetAsync(rowsum, 0, (size_t)HEADS * N * sizeof(float), stream);
  gat_edge_pass<<<(E + 255) / 256, 256, 0, stream>>>(src, dst, adj, S, e_ws, rowsum, N, E);

  // 4. per-head scatter + finalize
  hipMemsetAsync(out, 0, (size_t)N * FDIM * sizeof(float), stream);
  for (int k = 0; k < HEADS; ++k) {
    hipMemsetAsync(hbuf, 0, (size_t)N * FDIM * sizeof(float), stream);
    gat_scatter_head<<<(E + 7) / 8, 256, 0, stream>>>(src, dst, e_ws + (size_t)k * E,
                                                      X, hbuf, E);
    gat_finalize_head<<<(N * (FDIM / 4) + 255) / 256, 256, 0, stream>>>(
        hbuf, W + k * FDIM, rowsum + (size_t)k * N, out, N);
  }
}